// BezierHCPathOptimizer_63032940036318
// MI455X (gfx1250) — compile-verified
//
#include <hip/hip_runtime.h>
#include <math.h>

typedef __attribute__((ext_vector_type(2))) float v2f;
typedef __attribute__((ext_vector_type(8))) float v8f;

struct cpx { float x, y; };

__device__ __forceinline__ cpx cmul(cpx a, cpx b){ return {a.x*b.x - a.y*b.y, a.x*b.y + a.y*b.x}; }
__device__ __forceinline__ cpx csub(cpx a, cpx b){ return {a.x - b.x, a.y - b.y}; }
__device__ __forceinline__ cpx cdivq(cpx a, cpx b){
  float den = fmaxf(b.x*b.x + b.y*b.y, 1e-37f);
  float inv = 1.0f / den;
  return { (a.x*b.x + a.y*b.y)*inv, (a.y*b.x - a.x*b.y)*inv };
}
__device__ __forceinline__ float clogabs(cpx z){
  return 0.5f * logf(fmaxf(z.x*z.x + z.y*z.y, 1e-37f));
}
__device__ __forceinline__ float logaddexpf_(float a, float b){
  float mx = fmaxf(a, b);
  float mn = fminf(a, b);
  return mx + log1pf(expf(mn - mx));
}

// Euclidean resultant remainder sequence with log-magnitude accumulation.
// Pair (deg A = DB+1, deg B = DB). |Res(A,B)| = |lc B|^2 * |Res(B, R)|, R = A mod B,
// with R renormalized by |lc R| (contributes DB*log s). Terminal deg B == 0.
template<int DB>
__device__ __forceinline__ float res_steps(cpx* A, cpx* B, float acc){
  if constexpr (DB == 0) {
    return acc + clogabs(B[0]);
  } else {
    #pragma unroll
    for (int k = 0; k < 2; ++k){
      cpx q = cdivq(A[k], B[0]);
      #pragma unroll
      for (int j = 1; j <= DB; ++j) A[k + j] = csub(A[k + j], cmul(q, B[j]));
    }
    acc += 2.0f * clogabs(B[0]);
    float s2 = fmaxf(A[2].x*A[2].x + A[2].y*A[2].y, 1e-36f);
    acc += (float)DB * 0.5f * logf(s2);
    float inv = rsqrtf(s2);
    cpx R[DB];
    #pragma unroll
    for (int j = 0; j < DB; ++j){ R[j].x = A[2 + j].x * inv; R[j].y = A[2 + j].y * inv; }
    return res_steps<DB - 1>(B, R, acc);
  }
}

// Bernstein basis of degree D (D+1 values), zero-padded to 8.
template<int D>
__device__ __forceinline__ void bern(float t, float (&Bv)[8]){
  float u = 1.0f - t;
  float tp[8], up[8];
  tp[0] = 1.0f; up[0] = 1.0f;
  #pragma unroll
  for (int k = 1; k < 8; ++k){ tp[k] = tp[k-1]*t; up[k] = up[k-1]*u; }
  const float b7[8] = {1,7,21,35,35,21,7,1};
  const float b6[8] = {1,6,15,20,15,6,1,0};
  const float b5[8] = {1,5,10,10,5,1,0,0};
  #pragma unroll
  for (int k = 0; k < 8; ++k){
    float bb = (D == 7) ? b7[k] : (D == 6) ? b6[k] : b5[k];
    int ui = (k <= D) ? (D - k) : 0;
    Bv[k] = (k <= D) ? bb * tp[k] * up[ui] : 0.0f;
  }
}

// One wave evaluates a degree-D Bezier at its 32 samples via two 16x16x4 f32 WMMA
// tiles (K=8 in two chunks) and stages D (sample-major) into LDS.
// A layout (ISA 7.12.2): lane L -> row M=L&15; VGPR0 holds K=0|2, VGPR1 holds K=1|3
//   per half-wave. B layout assumed analogous: VGPR0 -> rows K=0|2, VGPR1 -> K=1|3.
template<int D>
__device__ __forceinline__ void wmma_eval(const float* __restrict__ ts,
                                          const float* __restrict__ sCe,
                                          float* __restrict__ stageW,
                                          int sampleBase, int lane, int M){
  const int col = lane & 15;
  const int hi  = (lane >> 4) & 1;
  #pragma unroll
  for (int tIdx = 0; tIdx < 2; ++tIdx){
    int srow = sampleBase + tIdx*16 + col;
    float tr = (srow < M) ? ts[srow] : 0.0f;
    float Bv[8]; bern<D>(tr, Bv);
    v8f d = {0.f,0.f,0.f,0.f,0.f,0.f,0.f,0.f};
    #pragma unroll
    for (int ch = 0; ch < 2; ++ch){
      float ax = hi ? Bv[ch*4 + 2] : Bv[ch*4 + 0];
      float ay = hi ? Bv[ch*4 + 3] : Bv[ch*4 + 1];
      v2f a; a.x = ax; a.y = ay;
      int k0 = ch*4 + hi*2;
      v2f b; b.x = sCe[k0*16 + col]; b.y = sCe[(k0 + 1)*16 + col];
      d = __builtin_amdgcn_wmma_f32_16x16x4_f32(false, a, false, b, (short)0, d, false, false);
    }
    #pragma unroll
    for (int r = 0; r < 8; ++r){
      stageW[(tIdx*16 + r + hi*8)*17 + col] = d[r];
    }
  }
}

__global__ __launch_bounds__(256)
void bez_main(const float* __restrict__ P0, const float* __restrict__ Pd,
              const float* __restrict__ Pmid, const float* __restrict__ ts,
              float* __restrict__ part, int M, int nblocks){
  __shared__ float sC[3*128];        // three 8x16 control matrices (deg7, deg6, deg5)
  __shared__ float stage[8*32*17];   // per-wave 32x16 staging (stride 17)
  __shared__ float red0[256], red1[256], red2[256];
  const int tid = threadIdx.x;

  // Build control matrices: C7 = P_ctrl, C6 = 7*dP (padded), C5 = 42*d2P (padded).
  for (int i = tid; i < 384; i += 256){
    int e = i >> 7, k = (i >> 4) & 7, N = i & 15;
    int n = N >> 1, cc = N & 1;
    auto pc = [&](int kk)->float {
      if (kk == 0) return P0[n*2 + cc];
      if (kk == 7) return Pd[n*2 + cc];
      return Pmid[((kk - 1)*8 + n)*2 + cc];
    };
    float v;
    if (e == 0)      v = pc(k);
    else if (e == 1) v = (k < 7) ? 7.0f*(pc(k+1) - pc(k)) : 0.0f;
    else             v = (k < 6) ? 42.0f*(pc(k+2) - 2.0f*pc(k+1) + pc(k)) : 0.0f;
    sC[i] = v;
  }
  __syncthreads();

  const int lane = tid & 31;
  const int wave = tid >> 5;
  const int sampleBase = blockIdx.x*256 + wave*32;
  const int mySample = sampleBase + lane;
  float* stageW = &stage[wave*32*17];

  // First derivative -> speed^2
  wmma_eval<6>(ts, sC + 128, stageW, sampleBase, lane, M);
  __syncthreads();
  float sp2 = 0.f;
  #pragma unroll
  for (int j = 0; j < 16; ++j){ float v = stageW[lane*17 + j]; sp2 += v*v; }
  __syncthreads();

  // Second derivative -> accel^2
  wmma_eval<5>(ts, sC + 256, stageW, sampleBase, lane, M);
  __syncthreads();
  float ac2 = 0.f;
  #pragma unroll
  for (int j = 0; j < 16; ++j){ float v = stageW[lane*17 + j]; ac2 += v*v; }
  __syncthreads();

  // Curve itself -> monic polynomial coefficients per sample
  wmma_eval<7>(ts, sC + 0, stageW, sampleBase, lane, M);
  __syncthreads();
  cpx F[9], G[8];
  F[0] = {1.0f, 0.0f};
  #pragma unroll
  for (int i = 0; i < 8; ++i){
    F[i+1].x = stageW[lane*17 + 2*i];
    F[i+1].y = stageW[lane*17 + 2*i + 1];
  }
  #pragma unroll
  for (int j = 0; j < 8; ++j){ float dk = (float)(8 - j); G[j].x = dk*F[j].x; G[j].y = dk*F[j].y; }

  // log|det Sylvester(f, f')| == log|Res(f, f')| via O(n^2) Euclid.
  float logabsdet = res_steps<7>(F, G, 0.0f);

  const float LOG_1EM12 = -27.631021115928547f;  // log(1e-12) == 2*log(1e-6)
  float disc = logaddexpf_(logabsdet, LOG_1EM12);              // - log1p(1e-12) == 0 in f32
  float log_softabs = 0.5f * logaddexpf_(2.0f*disc, LOG_1EM12);
  float lse = logaddexpf_(log_softabs, LOG_1EM12);
  float w = expf(-lse * 0.125f);
  float speed = sqrtf(sp2);

  float valid = (mySample < M) ? 1.0f : 0.0f;
  red0[tid] = valid * speed * w;
  red1[tid] = valid * sp2;
  red2[tid] = valid * ac2;
  __syncthreads();
  #pragma unroll
  for (int off = 128; off > 0; off >>= 1){
    if (tid < off){
      red0[tid] += red0[tid + off];
      red1[tid] += red1[tid + off];
      red2[tid] += red2[tid + off];
    }
    __syncthreads();
  }
  if (tid == 0){
    part[blockIdx.x]              = red0[0];
    part[nblocks + blockIdx.x]    = red1[0];
    part[2*nblocks + blockIdx.x]  = red2[0];
  }
}

__global__ __launch_bounds__(256)
void bez_finalize(const float* __restrict__ part, int nblocks, int M, float* __restrict__ out){
  __shared__ float r0[256], r1[256], r2[256];
  int tid = threadIdx.x;
  float s0 = 0.f, s1 = 0.f, s2 = 0.f;
  for (int i = tid; i < nblocks; i += 256){
    s0 += part[i];
    s1 += part[nblocks + i];
    s2 += part[2*nblocks + i];
  }
  r0[tid] = s0; r1[tid] = s1; r2[tid] = s2;
  __syncthreads();
  for (int off = 128; off > 0; off >>= 1){
    if (tid < off){ r0[tid] += r0[tid + off]; r1[tid] += r1[tid + off]; r2[tid] += r2[tid + off]; }
    __syncthreads();
  }
  if (tid == 0){
    float invM = 1.0f / (float)M;
    out[0] = r0[0]*invM + 0.1f*sqrtf(r1[0]*invM) + 0.01f*sqrtf(r2[0]*invM);
  }
}

extern "C" void kernel_launch(void* const* d_in, const int* in_sizes, int n_in,
                              void* d_out, int out_size, void* d_ws, size_t ws_size,
                              hipStream_t stream) {
  const float* P0   = (const float*)d_in[0];
  const float* Pd   = (const float*)d_in[1];
  const float* Pmid = (const float*)d_in[2];
  const float* ts   = (const float*)d_in[3];
  float* out  = (float*)d_out;
  float* part = (float*)d_ws;
  int M = in_sizes[3];
  int nblocks = (M + 255) / 256;
  bez_main<<<nblocks, 256, 0, stream>>>(P0, Pd, Pmid, ts, part, M, nblocks);
  bez_finalize<<<1, 256, 0, stream>>>(part, nblocks, M, out);
}